// GATv2Policy_84447646974221
// MI455X (gfx1250) — compile-verified
//
#include <hip/hip_runtime.h>

// GATv2 policy net on gfx1250 (MI455X).
// N=1024, HIDDEN=128, 3 layers, 4 heads x 32.
// - GEMMs (h@W and alpha@v) use V_WMMA_F32_16X16X4_F32 (fp32 matrix path).
// - The non-factorable leaky_relu pair tensor + softmax runs flash-style per
//   row, with DOUBLE-BUFFERED async global->LDS copies
//   (GLOBAL_LOAD_ASYNC_TO_LDS_B128, ASYNCcnt) hiding tile-fetch latency
//   behind the 32-deep VALU inner loop.

#define NN   1024
#define HID  128
#define NHEAD 4
#define HDIM 32
#define TPAD 132                 // padded LDS row stride (floats)
#define NEG_BIG (-1e9f)
#define LN_EPS  (1e-5f)

typedef float v2f __attribute__((ext_vector_type(2)));
typedef float v8f __attribute__((ext_vector_type(8)));
typedef int   v4i __attribute__((vector_size(16)));

__device__ __forceinline__ v8f wmma_f32_16x16x4(v2f a, v2f b, v8f c) {
  // 8 args: (neg_a, A, neg_b, B, c_mod, C, reuse_a, reuse_b)
  return __builtin_amdgcn_wmma_f32_16x16x4_f32(false, a, false, b, (short)0, c,
                                               false, false);
}

// ---- CDNA5 async global->LDS (ASYNCcnt) helpers ------------------------
// clang prints addrspace(1) as "__device__" in HIP diagnostics; the builtin's
// first param is v4i addrspace(1)* (global src), second is the LDS dest.
typedef __attribute__((address_space(1))) v4i* gv4i_p;
typedef __attribute__((address_space(3))) v4i* lv4i_p;

__device__ __forceinline__ void async_wait0() {
#if __has_builtin(__builtin_amdgcn_s_wait_asynccnt)
  __builtin_amdgcn_s_wait_asynccnt(0);
#else
  asm volatile("s_wait_asynccnt 0x0" ::: "memory");
#endif
}

// copy 16 bytes memory -> LDS, tracked by ASYNCcnt (no VGPR bounce)
__device__ __forceinline__ void async_copy16(const float* g, float* l) {
#if __has_builtin(__builtin_amdgcn_global_load_async_to_lds_b128)
  __builtin_amdgcn_global_load_async_to_lds_b128((gv4i_p)g, (lv4i_p)l, 0, 0);
#else
  unsigned loff = (unsigned)(unsigned long long)(lv4i_p)l;
  unsigned long long ga = (unsigned long long)(size_t)g;
  asm volatile("global_load_async_to_lds_b128 %0, %1, off"
               :: "v"(loff), "v"(ga) : "memory");
#endif
}

// -------- h0 = relu(nf @ Wp + bp) : (1024x3)@(3x128) --------------------
__global__ void k_proj(const float* __restrict__ nf, const float* __restrict__ Wp,
                       const float* __restrict__ bp, float* __restrict__ h) {
  int idx = blockIdx.x * blockDim.x + threadIdx.x;   // 0..131071
  int i = idx >> 7, c = idx & 127;
  float acc = bp[c];
#pragma unroll
  for (int k = 0; k < 3; k++) acc = fmaf(nf[i * 3 + k], Wp[k * HID + c], acc);
  h[idx] = fmaxf(acc, 0.f);
}

// -------- {wlh,wrh,v} = h @ {Wl,Wr,Wv} : 1024x128x128, WMMA f32 ---------
__global__ __launch_bounds__(32) void k_gemm_hw(
    const float* __restrict__ h,
    const float* __restrict__ W0, const float* __restrict__ W1,
    const float* __restrict__ W2,
    float* __restrict__ O0, float* __restrict__ O1, float* __restrict__ O2) {
  const float* W;
  float* O;
  if (blockIdx.z == 0) { W = W0; O = O0; }
  else if (blockIdx.z == 1) { W = W1; O = O1; }
  else { W = W2; O = O2; }

  int lane = threadIdx.x;
  int m0 = blockIdx.y * 16, n0 = blockIdx.x * 16;
  int mr = lane & 15, kh = lane >> 4;          // kh selects K-pair {0,1}/{2,3}

  v8f c = {0.f, 0.f, 0.f, 0.f, 0.f, 0.f, 0.f, 0.f};
  const float* arow = h + (m0 + mr) * HID + 2 * kh;        // A[mr][k+2kh..]
  const float* bcol = W + 2 * kh * HID + n0 + mr;          // B[k+2kh][mr]
  for (int k = 0; k < HID; k += 4) {
    v2f a, b;
    a.x = arow[k];
    a.y = arow[k + 1];
    b.x = bcol[k * HID];
    b.y = bcol[(k + 1) * HID];
    c = wmma_f32_16x16x4(a, b, c);
  }
  int mrow = m0 + 8 * kh;
#pragma unroll
  for (int r = 0; r < 8; r++) O[(mrow + r) * HID + n0 + mr] = c[r];
}

// -------- attention scores + masked softmax (per row i) -----------------
// e[i,j,h] = sum_d att[d]*leaky(wlh[i,h,d]+wrh[j,h,d]); alpha = softmax_j.
// 256 threads/row: head = t>>6, j-slot = t&63. wrh staged in double-buffered
// LDS tiles (64 rows, padded stride 132) via async global->LDS b128 copies;
// tile p+1 is in flight while tile p is consumed.
__global__ __launch_bounds__(256) void k_attn(
    const float* __restrict__ wlh, const float* __restrict__ wrh,
    const float* __restrict__ adj, const float* __restrict__ att,
    float* __restrict__ alpha) {
  __shared__ float tile[2][64 * TPAD];
  __shared__ float att_s[HDIM];
  __shared__ float red[256];

  int i = blockIdx.x;
  int t = threadIdx.x;
  int hh = t >> 6;
  int jb = t & 63;
  if (t < HDIM) att_s[t] = att[t];

  float wl[HDIM];
#pragma unroll
  for (int d = 0; d < HDIM; d++) wl[d] = wlh[i * HID + hh * HDIM + d];

  // issue async copies for tile 0 (8 x b128 per thread = 64x128 floats)
#pragma unroll
  for (int q = 0; q < 8; q++) {
    int idx = t + q * 256;
    int row = idx >> 5, c4 = idx & 31;
    async_copy16(wrh + row * HID + c4 * 4, &tile[0][row * TPAD + c4 * 4]);
  }

  float ev[16];
#pragma unroll
  for (int p = 0; p < 16; p++) {
    async_wait0();        // this wave's copies done
    __syncthreads();      // all waves' copies visible / prev tile consumed
    if (p < 15) {
      const float* src = wrh + (p + 1) * 64 * HID;
      float* dbuf = &tile[(p + 1) & 1][0];
#pragma unroll
      for (int q = 0; q < 8; q++) {
        int idx = t + q * 256;
        int row = idx >> 5, c4 = idx & 31;
        async_copy16(src + row * HID + c4 * 4, dbuf + row * TPAD + c4 * 4);
      }
    }
    float e = 0.f;
    const float* wr = &tile[p & 1][jb * TPAD + hh * HDIM];
#pragma unroll
    for (int d = 0; d < HDIM; d++) {
      float x = wl[d] + wr[d];
      e = fmaf(att_s[d], fmaxf(x, 0.2f * x), e);   // leaky_relu(x,0.2)
    }
    int j = p * 64 + jb;
    ev[p] = (adj[i * NN + j] > 0.5f) ? e : NEG_BIG;
  }

  // per-head softmax over all 1024 j (64 threads x 16 values per head)
  float mloc = ev[0];
#pragma unroll
  for (int p = 1; p < 16; p++) mloc = fmaxf(mloc, ev[p]);
  red[t] = mloc;
  __syncthreads();
  float mh = red[hh * 64];
  for (int q = 1; q < 64; q++) mh = fmaxf(mh, red[hh * 64 + q]);
  __syncthreads();
  float sloc = 0.f;
#pragma unroll
  for (int p = 0; p < 16; p++) {
    ev[p] = __expf(ev[p] - mh);
    sloc += ev[p];
  }
  red[t] = sloc;
  __syncthreads();
  float sh = 0.f;
  for (int q = 0; q < 64; q++) sh += red[hh * 64 + q];
  float inv = 1.0f / sh;

  float* arow = alpha + (size_t)(hh * NN + i) * NN;
#pragma unroll
  for (int p = 0; p < 16; p++) arow[p * 64 + jb] = ev[p] * inv;
}

// -------- agg[:,h,:] = alpha_h (1024x1024) @ v_h (1024x32), WMMA f32 ----
__global__ __launch_bounds__(32) void k_agg(const float* __restrict__ alpha,
                                            const float* __restrict__ v,
                                            float* __restrict__ agg) {
  int lane = threadIdx.x;
  int hh = blockIdx.z;
  int m0 = blockIdx.y * 16, n0 = blockIdx.x * 16;
  int mr = lane & 15, kh = lane >> 4;

  v8f c = {0.f, 0.f, 0.f, 0.f, 0.f, 0.f, 0.f, 0.f};
  const float* arow = alpha + (size_t)(hh * NN + m0 + mr) * NN + 2 * kh;
  const float* bcol = v + hh * HDIM + n0 + mr + 2 * kh * HID;
  for (int k = 0; k < NN; k += 4) {
    v2f a, b;
    a.x = arow[k];
    a.y = arow[k + 1];
    b.x = bcol[k * HID];
    b.y = bcol[(k + 1) * HID];
    c = wmma_f32_16x16x4(a, b, c);
  }
  int mrow = m0 + 8 * kh;
#pragma unroll
  for (int r = 0; r < 8; r++)
    agg[(mrow + r) * HID + hh * HDIM + n0 + mr] = c[r];
}

// -------- h += relu(LN(agg)*gamma + beta) -------------------------------
__global__ __launch_bounds__(128) void k_lnres(const float* __restrict__ agg,
                                               const float* __restrict__ gamma,
                                               const float* __restrict__ beta,
                                               float* __restrict__ h) {
  __shared__ float s1[HID], s2[HID];
  int i = blockIdx.x, c = threadIdx.x;
  float a = agg[i * HID + c];
  s1[c] = a;
  s2[c] = a * a;
  __syncthreads();
  float s = 0.f, ss = 0.f;
  for (int q = 0; q < HID; q++) { s += s1[q]; ss += s2[q]; }
  float mu = s * (1.f / HID);
  float var = ss * (1.f / HID) - mu * mu;          // population var (ddof=0)
  float o = (a - mu) * rsqrtf(var + LN_EPS) * gamma[c] + beta[c];
  h[i * HID + c] += fmaxf(o, 0.f);
}

// -------- node_logits + delta_mu heads ----------------------------------
__global__ __launch_bounds__(64) void k_heads(
    const float* __restrict__ h, const float* __restrict__ Wn1,
    const float* __restrict__ bn1, const float* __restrict__ Wn2,
    const float* __restrict__ bn2, const float* __restrict__ Wd1,
    const float* __restrict__ bd1, const float* __restrict__ Wd2,
    const float* __restrict__ bd2, float* __restrict__ out) {
  __shared__ float r0[64], r1[64], r2[64];
  int i = blockIdx.x, j = threadIdx.x;
  float a1 = 0.f, a2 = 0.f;
  for (int c = 0; c < HID; c++) {
    float hv = h[i * HID + c];
    a1 = fmaf(hv, Wn1[c * 64 + j], a1);
    a2 = fmaf(hv, Wd1[c * 64 + j], a2);
  }
  float t1 = fmaxf(a1 + bn1[j], 0.f);
  float t2 = fmaxf(a2 + bd1[j], 0.f);
  r0[j] = t1 * Wn2[j];
  r1[j] = t2 * Wd2[2 * j];
  r2[j] = t2 * Wd2[2 * j + 1];
  __syncthreads();
  if (j == 0) {
    float s0 = 0.f, sa = 0.f, sb = 0.f;
    for (int q = 0; q < 64; q++) { s0 += r0[q]; sa += r1[q]; sb += r2[q]; }
    out[i] = s0 + bn2[0];                     // node_logits
    out[NN + 2 * i]     = sa + bd2[0];        // delta_mu[:,0]
    out[NN + 2 * i + 1] = sb + bd2[1];        // delta_mu[:,1]
  }
}

// -------- pooled mean + value head --------------------------------------
__global__ __launch_bounds__(128) void k_value(
    const float* __restrict__ h, const float* __restrict__ Wv1,
    const float* __restrict__ bv1, const float* __restrict__ Wv2,
    const float* __restrict__ bv2, float* __restrict__ out) {
  __shared__ float pool[HID];
  __shared__ float r[64];
  int t = threadIdx.x;
  float s = 0.f;
  for (int i = 0; i < NN; i++) s += h[i * HID + t];
  pool[t] = s * (1.f / NN);
  __syncthreads();
  if (t < 64) {
    float a = 0.f;
    for (int c = 0; c < HID; c++) a = fmaf(pool[c], Wv1[c * 64 + t], a);
    r[t] = fmaxf(a + bv1[t], 0.f) * Wv2[t];
  }
  __syncthreads();
  if (t == 0) {
    float sv = 0.f;
    for (int q = 0; q < 64; q++) sv += r[q];
    out[3 * NN] = sv + bv2[0];                // value at out[3072]
  }
}

extern "C" void kernel_launch(void* const* d_in, const int* in_sizes, int n_in,
                              void* d_out, int out_size, void* d_ws,
                              size_t ws_size, hipStream_t stream) {
  const float* nf    = (const float*)d_in[0];
  const float* adj   = (const float*)d_in[1];
  const float* Wp    = (const float*)d_in[2];
  const float* bp    = (const float*)d_in[3];
  const float* Wl    = (const float*)d_in[4];
  const float* Wr    = (const float*)d_in[5];
  const float* Wv    = (const float*)d_in[6];
  const float* att   = (const float*)d_in[7];
  const float* gamma = (const float*)d_in[8];
  const float* beta  = (const float*)d_in[9];
  const float* Wn1 = (const float*)d_in[10]; const float* bn1 = (const float*)d_in[11];
  const float* Wn2 = (const float*)d_in[12]; const float* bn2 = (const float*)d_in[13];
  const float* Wd1 = (const float*)d_in[14]; const float* bd1 = (const float*)d_in[15];
  const float* Wd2 = (const float*)d_in[16]; const float* bd2 = (const float*)d_in[17];
  const float* Wv1 = (const float*)d_in[18]; const float* bv1 = (const float*)d_in[19];
  const float* Wv2 = (const float*)d_in[20]; const float* bv2 = (const float*)d_in[21];
  float* out = (float*)d_out;

  // Workspace layout (floats): ~18.5 MB total, L2-resident on MI455X.
  float* h     = (float*)d_ws;           // 1024*128
  float* wlh   = h     + NN * HID;       // 1024*128
  float* wrh   = wlh   + NN * HID;       // 1024*128
  float* vv    = wrh   + NN * HID;       // 1024*128
  float* agg   = vv    + NN * HID;       // 1024*128
  float* alpha = agg   + NN * HID;       // 4 * 1024 * 1024

  k_proj<<<(NN * HID) / 256, 256, 0, stream>>>(nf, Wp, bp, h);
  for (int l = 0; l < 3; l++) {
    k_gemm_hw<<<dim3(HID / 16, NN / 16, 3), 32, 0, stream>>>(
        h, Wl + l * HID * HID, Wr + l * HID * HID, Wv + l * HID * HID,
        wlh, wrh, vv);
    k_attn<<<NN, 256, 0, stream>>>(wlh, wrh, adj, att + l * HDIM, alpha);
    k_agg<<<dim3(HDIM / 16, NN / 16, NHEAD), 32, 0, stream>>>(alpha, vv, agg);
    k_lnres<<<NN, HID, 0, stream>>>(agg, gamma + l * HID, beta + l * HID, h);
  }
  k_heads<<<NN, 64, 0, stream>>>(h, Wn1, bn1, Wn2, bn2, Wd1, bd1, Wd2, bd2, out);
  k_value<<<1, HID, 0, stream>>>(h, Wv1, bv1, Wv2, bv2, out);
}